// TypeAttention_27470610825502
// MI455X (gfx1250) — compile-verified
//
#include <hip/hip_runtime.h>

#define DIM   128          // hidden dim D
#define LDSS  132          // padded LDS row stride (conflict-free for 16x4 fragments)
#define SLOPE 0.2f

typedef float v2f __attribute__((ext_vector_type(2)));
typedef float v8f __attribute__((ext_vector_type(8)));

// ---------------------------------------------------------------- utilities
__global__ void zero_f32(float* __restrict__ p, long n) {
  long i = (long)blockIdx.x * blockDim.x + threadIdx.x;
  long stride = (long)gridDim.x * blockDim.x;
  for (; i < n; i += stride) p[i] = 0.0f;
}

// --------------------------- pack W (row-major KxN) into WMMA fragment order
// Wp[((t*DIM + c)*2 + h)*2 + j] = W[(4t + 2h + j)*DIM + c]
// so a lane's B-fragment {W[kk][c], W[kk+1][c]} (kk = 4t + 2h) is one float2.
__global__ void pack_w_kernel(const float* __restrict__ W, float* __restrict__ Wp) {
  int i = blockIdx.x * blockDim.x + threadIdx.x;
  if (i >= DIM * DIM) return;
  int j = i & 1;
  int h = (i >> 1) & 1;
  int c = (i >> 2) & (DIM - 1);
  int t = i >> 9;
  Wp[i] = W[(4 * t + 2 * h + j) * DIM + c];
}

// ------------------------------------------------------- degree histograms
__global__ void degrees_kernel(const int* __restrict__ src, const int* __restrict__ dst,
                               float* __restrict__ outdeg, float* __restrict__ indeg,
                               int E) {
  int e = blockIdx.x * blockDim.x + threadIdx.x;
  if (e < E) {
    atomicAdd(&outdeg[src[e]], 1.0f);
    atomicAdd(&indeg[dst[e]], 1.0f);
  }
}

// --------------------------- rst[dst] += h_src[src] * outdeg[src]^-1/2
// one wave (32 lanes) per edge, float4 per lane = 128 floats
__global__ void scatter_kernel(const float* __restrict__ hsrc, const int* __restrict__ src,
                               const int* __restrict__ dst, const float* __restrict__ outdeg,
                               float* __restrict__ rst, int E) {
  int e    = blockIdx.x * (blockDim.x >> 5) + (threadIdx.x >> 5);
  int lane = threadIdx.x & 31;
  if (e >= E) return;
  int s = src[e];
  int d = dst[e];
  float sc = rsqrtf(fmaxf(outdeg[s], 1.0f));
  const float4 h = *(const float4*)(hsrc + (long)s * DIM + lane * 4);
  float* base = rst + (long)d * DIM + lane * 4;
  atomicAdd(base + 0, h.x * sc);
  atomicAdd(base + 1, h.y * sc);
  atomicAdd(base + 2, h.z * sc);
  atomicAdd(base + 3, h.w * sc);
}

// ------------- m = elu(h_dst @ Wl + bl + (rst * indeg^-1/2) @ Wr + br)
// 256 threads = 8 waves. Block tile: 16 rows x 128 cols; wave w owns the
// 16x16 column tile w. A tiles staged in LDS (shared by all 8 waves),
// fp32 WMMA 16x16x4, bias folded into the accumulator, B from packed W
// (single b64 load per fragment).
__global__ __launch_bounds__(256) void gemm_elu_wmma_kernel(
    const float* __restrict__ hdst, const float* __restrict__ rst,
    const float* __restrict__ indeg,
    const float* __restrict__ Wpl, const float* __restrict__ bl,
    const float* __restrict__ Wpr, const float* __restrict__ br,
    float* __restrict__ mbuf, int N) {
  __shared__ float sA0[16 * LDSS];   // h_dst tile
  __shared__ float sA1[16 * LDSS];   // rst tile, pre-scaled by indeg^-1/2

  const int rowbase = blockIdx.x * 16;
  const int t = threadIdx.x;

  // ---- cooperative stage: thread t loads 8 floats of row (t>>4)
  {
    int srow = t >> 4;
    int scol = (t & 15) * 8;
    int grow = rowbase + srow;
    if (grow >= N) grow = N - 1;               // clamp (no divergence)
    long g = (long)grow * DIM + scol;
    float insc = rsqrtf(fmaxf(indeg[grow], 1.0f));
    float4 h0 = *(const float4*)(hdst + g);
    float4 h1 = *(const float4*)(hdst + g + 4);
    float4 r0 = *(const float4*)(rst + g);
    float4 r1 = *(const float4*)(rst + g + 4);
    r0.x *= insc; r0.y *= insc; r0.z *= insc; r0.w *= insc;
    r1.x *= insc; r1.y *= insc; r1.z *= insc; r1.w *= insc;
    *(float4*)(sA0 + srow * LDSS + scol)     = h0;
    *(float4*)(sA0 + srow * LDSS + scol + 4) = h1;
    *(float4*)(sA1 + srow * LDSS + scol)     = r0;
    *(float4*)(sA1 + srow * LDSS + scol + 4) = r1;
  }
  __syncthreads();

  const int wave  = t >> 5;
  const int lane  = t & 31;
  const int lr    = lane & 15;          // A: row M, B/C: col N
  const int khalf = (lane >> 4) * 2;    // K sub-offset per ISA fragment layout
  const int colb  = wave * 16;

  // per-lane base into packed W; per k-step stride is DIM*4 = 512 floats
  const int boff = (((colb + lr) * 2) + (lane >> 4)) * 2;
  const float* pWl = Wpl + boff;
  const float* pWr = Wpr + boff;

  float bv = bl[colb + lr] + br[colb + lr];
  v8f acc;
#pragma unroll
  for (int i = 0; i < 8; ++i) acc[i] = bv;

#pragma unroll 8
  for (int ks = 0; ks < DIM / 4; ++ks) {
    int kk = ks * 4 + khalf;
    v2f a1 = { sA0[lr * LDSS + kk], sA0[lr * LDSS + kk + 1] };
    v2f a2 = { sA1[lr * LDSS + kk], sA1[lr * LDSS + kk + 1] };
    v2f b1 = *(const v2f*)(pWl + ks * (DIM * 4));
    v2f b2 = *(const v2f*)(pWr + ks * (DIM * 4));
    acc = __builtin_amdgcn_wmma_f32_16x16x4_f32(false, a1, false, b1,
                                                (short)0, acc, false, false);
    acc = __builtin_amdgcn_wmma_f32_16x16x4_f32(false, a2, false, b2,
                                                (short)0, acc, false, false);
  }

  // ---- epilogue: ELU + store. C/D layout: VGPR i holds row rowbase+i
  // (lanes 0-15) or rowbase+i+8 (lanes 16-31), column = colb + (lane&15).
  int rbase2 = rowbase + ((lane >> 4) << 3);
  int col = colb + lr;
  if (rowbase + 16 <= N) {               // block-uniform fast path (scalar branch)
#pragma unroll
    for (int i = 0; i < 8; ++i) {
      float v = acc[i];
      v = (v > 0.0f) ? v : (__expf(v) - 1.0f);     // elu, alpha=1
      mbuf[(long)(rbase2 + i) * DIM + col] = v;
    }
  } else {
#pragma unroll
    for (int i = 0; i < 8; ++i) {
      int r = rbase2 + i;
      if (r < N) {
        float v = acc[i];
        v = (v > 0.0f) ? v : (__expf(v) - 1.0f);
        mbuf[(long)r * DIM + col] = v;
      }
    }
  }
}

// --------------------------- logit = leaky_relu(m @ attn_w), one wave/row
__global__ void logit_kernel(const float* __restrict__ mbuf, const float* __restrict__ aw,
                             float* __restrict__ logit, int N) {
  int r    = blockIdx.x * (blockDim.x >> 5) + (threadIdx.x >> 5);
  int lane = threadIdx.x & 31;
  if (r >= N) return;
  float4 mv = *(const float4*)(mbuf + (long)r * DIM + lane * 4);
  float4 av = *(const float4*)(aw + lane * 4);
  float s = mv.x * av.x + mv.y * av.y + mv.z * av.z + mv.w * av.w;
#pragma unroll
  for (int off = 16; off > 0; off >>= 1) s += __shfl_xor(s, off, 32);
  if (lane == 0) logit[r] = (s > 0.0f) ? s : SLOPE * s;
}

// ---- edge softmax. emax[d] == logit[d] exactly (all incoming edges of a dst
// share one logit), so the max pass is the identity; denom via atomics.
__global__ void denom_kernel(const int* __restrict__ dst, const float* __restrict__ logit,
                             float* __restrict__ denom, int E) {
  int e = blockIdx.x * blockDim.x + threadIdx.x;
  if (e < E) {
    int d = dst[e];
    float lg = logit[d];
    atomicAdd(&denom[d], __expf(lg - lg));
  }
}

__global__ void alpha_kernel(const int* __restrict__ dst, const float* __restrict__ logit,
                             const float* __restrict__ denom, float* __restrict__ out, int E) {
  int e = blockIdx.x * blockDim.x + threadIdx.x;
  if (e < E) {
    int d = dst[e];
    float lg = logit[d];
    out[e] = __expf(lg - lg) / denom[d];
  }
}

// ---------------------------------------------------------------- driver
static void run_relation(const float* h_src, const float* h_dst,
                         const float* Wl, const float* bl,
                         const float* Wr, const float* br,
                         const float* attn_w,
                         const int* src, const int* dst,
                         int N, int E,
                         float* rst, float* mbuf, float* outdeg, float* indeg,
                         float* logit, float* denom,
                         float* wpl, float* wpr,
                         float* out_alpha, hipStream_t stream) {
  // zero rst + (outdeg,indeg,logit,denom) region
  zero_f32<<<1024, 256, 0, stream>>>(rst, (long)N * DIM);
  zero_f32<<<1024, 256, 0, stream>>>(outdeg, 4L * N);

  pack_w_kernel<<<(DIM * DIM + 255) / 256, 256, 0, stream>>>(Wl, wpl);
  pack_w_kernel<<<(DIM * DIM + 255) / 256, 256, 0, stream>>>(Wr, wpr);

  degrees_kernel<<<(E + 255) / 256, 256, 0, stream>>>(src, dst, outdeg, indeg, E);
  scatter_kernel<<<(E + 7) / 8, 256, 0, stream>>>(h_src, src, dst, outdeg, rst, E);
  gemm_elu_wmma_kernel<<<(N + 15) / 16, 256, 0, stream>>>(
      h_dst, rst, indeg, wpl, bl, wpr, br, mbuf, N);
  logit_kernel<<<(N + 7) / 8, 256, 0, stream>>>(mbuf, attn_w, logit, N);
  denom_kernel<<<(E + 255) / 256, 256, 0, stream>>>(dst, logit, denom, E);
  alpha_kernel<<<(E + 255) / 256, 256, 0, stream>>>(dst, logit, denom, out_alpha, E);
}

extern "C" void kernel_launch(void* const* d_in, const int* in_sizes, int n_in,
                              void* d_out, int out_size, void* d_ws, size_t ws_size,
                              hipStream_t stream) {
  const float* h_user  = (const float*)d_in[0];
  const float* h_item  = (const float*)d_in[1];
  const float* Wl_user = (const float*)d_in[2];
  const float* bl_user = (const float*)d_in[3];
  const float* Wl_item = (const float*)d_in[4];
  const float* bl_item = (const float*)d_in[5];
  const float* Wr_user = (const float*)d_in[6];
  const float* br_user = (const float*)d_in[7];
  const float* Wr_item = (const float*)d_in[8];
  const float* br_item = (const float*)d_in[9];
  const float* attn_w  = (const float*)d_in[10];
  const int*   src_ui  = (const int*)d_in[11];
  const int*   dst_ui  = (const int*)d_in[12];
  const int*   src_iu  = (const int*)d_in[13];
  const int*   dst_iu  = (const int*)d_in[14];

  const int N = in_sizes[0] / DIM;
  const int E = in_sizes[11];

  float* ws     = (float*)d_ws;
  float* rst    = ws;                          // N*D
  float* mbuf   = rst + (size_t)N * DIM;       // N*D
  float* outdeg = mbuf + (size_t)N * DIM;      // N
  float* indeg  = outdeg + N;                  // N
  float* logit  = indeg + N;                   // N
  float* denom  = logit + N;                   // N
  float* wpl    = denom + N;                   // D*D packed Wl
  float* wpr    = wpl + DIM * DIM;             // D*D packed Wr
  (void)ws_size; (void)n_in; (void)out_size;

  float* out = (float*)d_out;

  // relation user->item (dsttype = item)
  run_relation(h_user, h_item, Wl_item, bl_item, Wr_item, br_item, attn_w,
               src_ui, dst_ui, N, E,
               rst, mbuf, outdeg, indeg, logit, denom, wpl, wpr, out, stream);
  // relation item->user (dsttype = user)
  run_relation(h_item, h_user, Wl_user, bl_user, Wr_user, br_user, attn_w,
               src_iu, dst_iu, N, E,
               rst, mbuf, outdeg, indeg, logit, denom, wpl, wpr, out + E, stream);
}